// DeepSetEquivariant_18519898980591
// MI455X (gfx1250) — compile-verified
//
#include <hip/hip_runtime.h>

typedef float v2f __attribute__((ext_vector_type(2)));
typedef float v8f __attribute__((ext_vector_type(8)));

#define D 128

// ---------------------------------------------------------------------------
// Kernel 1: column sum of x (pooled[j] = sum_i x[i][j]) with float4 loads.
// 256 threads: 32 float4-columns x 8 rows per iteration. Also pre-warms x
// into the 192MB L2 so the GEMM pass re-reads it from L2, not HBM.
// ---------------------------------------------------------------------------
__global__ __launch_bounds__(256) void colsum_kernel(const float4* __restrict__ x4,
                                                     float* __restrict__ pooled,
                                                     int n) {
    const int tid = threadIdx.x;
    const int c4  = tid & 31;                       // float4 column 0..31
    const int rowStart = blockIdx.x * 8 + (tid >> 5);
    const int rowStep  = gridDim.x * 8;
    float4 s = make_float4(0.f, 0.f, 0.f, 0.f);
    for (int r = rowStart; r < n; r += rowStep) {
        float4 v = x4[(size_t)r * 32 + c4];
        s.x += v.x; s.y += v.y; s.z += v.z; s.w += v.w;
    }
    __shared__ float4 red[256];
    red[tid] = s;
    __syncthreads();
    if (tid < 32) {
        float4 t = red[tid];
        #pragma unroll
        for (int g = 1; g < 8; ++g) {
            float4 v = red[g * 32 + tid];
            t.x += v.x; t.y += v.y; t.z += v.z; t.w += v.w;
        }
        atomicAdd(&pooled[4 * tid + 0], t.x);
        atomicAdd(&pooled[4 * tid + 1], t.y);
        atomicAdd(&pooled[4 * tid + 2], t.z);
        atomicAdd(&pooled[4 * tid + 3], t.w);
    }
}

// ---------------------------------------------------------------------------
// Kernel 2: tvec[j] = (pooled @ w2)[j] / n + bias[j]   (tiny, one block)
// ---------------------------------------------------------------------------
__global__ __launch_bounds__(128) void finalize_kernel(const float* __restrict__ pooled,
                                                       const float* __restrict__ w2,
                                                       const float* __restrict__ bias,
                                                       float* __restrict__ tvec,
                                                       float inv_n) {
    const int j = threadIdx.x;
    float s = 0.0f;
    #pragma unroll 8
    for (int k = 0; k < D; ++k) {
        s += pooled[k] * w2[k * D + j];
    }
    tvec[j] = s * inv_n + bias[j];
}

// ---------------------------------------------------------------------------
// Kernel 3: out[i][j] = (x @ w1)[i][j] + tvec[j]  via V_WMMA_F32_16X16X4_F32.
// Block = 256 threads = 8 waves; each wave computes a 16x128 output strip.
// 64KB LDS is used twice: first as the packed w1 B-fragment buffer, then
// (after a block-uniform barrier) as a store-transpose buffer so the epilogue
// is 16 fully-coalesced global_store_b128 per lane (512B per wave-instr).
// ---------------------------------------------------------------------------
__global__ __launch_bounds__(256) void gemm_kernel(const float* __restrict__ x,
                                                   const float* __restrict__ w1,
                                                   const float* __restrict__ tvec,
                                                   float* __restrict__ out,
                                                   int n) {
    __shared__ float smem[16384];        // 64 KB, dual-purpose
    v2f* w1p = (v2f*)smem;               // [32][128][2] packed B fragments

    const int tid = threadIdx.x;

    // Cooperative one-time pack of w1 into LDS (B-fragment layout):
    //   w1p[(kk*128 + col)*2 + g] = { w1[4kk+2g][col], w1[4kk+2g+1][col] }
    for (int idx = tid; idx < 32 * 128 * 2; idx += 256) {
        const int g   = idx & 1;
        const int col = (idx >> 1) & (D - 1);
        const int kk  = idx >> 8;
        const int k0  = 4 * kk + 2 * g;
        v2f v;
        v.x = w1[(k0 + 0) * D + col];
        v.y = w1[(k0 + 1) * D + col];
        w1p[idx] = v;
    }

    const int wave = tid >> 5;
    const int lane = tid & 31;
    const int g    = lane >> 4;          // half-wave: K-pair select
    const int l15  = lane & 15;

    const long R = (long)blockIdx.x * 128 + (long)wave * 16;  // strip row base

    // Broadcast transmit values for this lane's output columns.
    float tv[8];
    #pragma unroll
    for (int t = 0; t < 8; ++t) tv[t] = tvec[t * 16 + l15];

    __syncthreads();

    v8f acc[8] = {};

    // A fragment row for this lane (clamped for the n tail; stores are guarded
    // there, and the hot path only runs on full blocks).
    long rowA = R + l15;
    if (rowA >= n) rowA = n - 1;
    const float* xrow = x + rowA * D + 2 * g;

    for (int kk = 0; kk < 32; ++kk) {
        // A 16x4 fragment: lanes 0-15 get K = 4kk+{0,1}, lanes 16-31 K = 4kk+{2,3}
        const v2f a = *(const v2f*)(xrow + 4 * kk);
        const v2f* bbase = &w1p[(kk * 128) * 2 + g];
        #pragma unroll
        for (int t = 0; t < 8; ++t) {
            const v2f b = bbase[(t * 16 + l15) * 2];
            acc[t] = __builtin_amdgcn_wmma_f32_16x16x4_f32(
                /*neg_a=*/false, a, /*neg_b=*/false, b,
                /*c_mod=*/(short)0, acc[t],
                /*reuse_a=*/false, /*reuse_b=*/false);
        }
    }

    // ---- Epilogue ----------------------------------------------------------
    // Block-uniform branch: every full block takes the coalesced b128 path.
    const bool fullBlock = ((long)(blockIdx.x + 1) * 128) <= (long)n;
    if (fullBlock) {
        __syncthreads();                 // all waves done reading w1p
        float* tbuf = smem + wave * (16 * D);   // this wave's 8KB region
        // Scatter C fragments (+ transmit) into row-major strip layout.
        #pragma unroll
        for (int t = 0; t < 8; ++t) {
            const int col = t * 16 + l15;
            #pragma unroll
            for (int r = 0; r < 8; ++r) {
                tbuf[(r + 8 * g) * D + col] = acc[t][r] + tv[t];
            }
        }
        __syncthreads();                 // transpose visible to whole wave
        const float4* tb4 = (const float4*)tbuf;
        // 16 rows, one full 512B row per wave-instruction.
        #pragma unroll
        for (int i = 0; i < 16; ++i) {
            *(float4*)(out + (R + i) * D + 4 * lane) = tb4[i * 32 + lane];
        }
    } else {
        // Tail block (at most one per grid): guarded scalar stores.
        const long rowBase = R + (long)g * 8;
        #pragma unroll
        for (int t = 0; t < 8; ++t) {
            const int col = t * 16 + l15;
            #pragma unroll
            for (int r = 0; r < 8; ++r) {
                const long row = rowBase + r;
                if (row < n) {
                    out[row * D + col] = acc[t][r] + tv[t];
                }
            }
        }
    }
}

// ---------------------------------------------------------------------------
extern "C" void kernel_launch(void* const* d_in, const int* in_sizes, int n_in,
                              void* d_out, int out_size, void* d_ws, size_t ws_size,
                              hipStream_t stream) {
    const float* x    = (const float*)d_in[0];
    const float* w1   = (const float*)d_in[1];
    const float* w2   = (const float*)d_in[2];
    const float* bias = (const float*)d_in[3];
    float* out = (float*)d_out;

    const int n = in_sizes[0] / D;

    float* pooled = (float*)d_ws;       // 128 floats
    float* tvec   = pooled + D;         // 128 floats

    hipMemsetAsync(pooled, 0, D * sizeof(float), stream);

    colsum_kernel<<<1024, 256, 0, stream>>>((const float4*)x, pooled, n);
    finalize_kernel<<<1, 128, 0, stream>>>(pooled, w2, bias, tvec, 1.0f / (float)n);

    const int nblocks = (n + 127) / 128;
    gemm_kernel<<<nblocks, 256, 0, stream>>>(x, w1, tvec, out, n);
}